// Network_65326452572674
// MI455X (gfx1250) — compile-verified
//
#include <hip/hip_runtime.h>
#include <hip/hip_bf16.h>
#include <math.h>

// Problem constants (match reference)
#define B_SZ    64
#define IN_DIM  1024
#define OUT_DIM 512
#define NN      2048
#define LEAKY   0.01f

typedef __attribute__((ext_vector_type(2))) float v2f;
typedef __attribute__((ext_vector_type(8))) float v8f;
typedef unsigned char u8;

// ---------------------------------------------------------------------------
// Nonlinearity dispatch: 0 ReLU, 1 Tanh, 2 LeakyReLU, 3 Identity, 4 Sine,
// 5 Softplus, 6 SoftMinus  (nl_idx in [0,6])
// ---------------------------------------------------------------------------
__device__ __forceinline__ float apply_nl(int idx, float v) {
    switch (idx) {
        case 0:  return v > 0.f ? v : 0.f;
        case 1:  return tanhf(v);
        case 2:  return v >= 0.f ? v : LEAKY * v;
        case 3:  return v;
        case 4:  return sinf(v);
        case 5:  return v > 20.f ? v : log1pf(expf(v));          // softplus
        default: return v < -20.f ? v : -log1pf(expf(-v));      // -softplus(-v)
    }
}

// ---------------------------------------------------------------------------
// Kernel 1: masked transpose of W_mid -> Wt[n,k] = W_mid[k,n] * adj[k,n]
// LDS-tiled so both read and write are coalesced. One-time ~20 MB of traffic;
// leaves Wt resident in the 192 MB L2 for the scan.
// ---------------------------------------------------------------------------
__global__ __launch_bounds__(256) void tmask_kernel(const float* __restrict__ Wm,
                                                    const u8* __restrict__ adj,
                                                    float* __restrict__ Wt) {
    __shared__ float tile[32][33];
    const int n0 = blockIdx.x * 32;   // column tile of W_mid
    const int k0 = blockIdx.y * 32;   // row tile of W_mid
    const int tx = threadIdx.x;       // 0..31
    const int ty = threadIdx.y;       // 0..7
    #pragma unroll
    for (int i = ty; i < 32; i += 8) {
        const int k = k0 + i, n = n0 + tx;
        const float v = Wm[k * NN + n];
        tile[i][tx] = adj[k * NN + n] ? v : 0.f;
    }
    __syncthreads();
    #pragma unroll
    for (int i = ty; i < 32; i += 8) {
        const int n = n0 + i, k = k0 + tx;
        Wt[n * NN + k] = tile[tx][i];
    }
}

// ---------------------------------------------------------------------------
// Kernel 2: mcol[k] = adj[k, N-1] ? 1 : 0  (for the output GEMM mask)
// ---------------------------------------------------------------------------
__global__ void maskcol_kernel(const u8* __restrict__ adj, float* __restrict__ mcol) {
    const int k = blockIdx.x * 256 + threadIdx.x;
    if (k < NN) mcol[k] = adj[k * NN + (NN - 1)] ? 1.f : 0.f;
}

// ---------------------------------------------------------------------------
// Kernel 3: GEMM0  state = mask(adj[0]) * nl0(X @ W0 + b0)
// Full-precision WMMA: V_WMMA_F32_16X16X4_F32, one wave per 16x16 C tile.
// A 16x4 layout: lane -> M (0..15 per half), VGPR0 = K=2h, VGPR1 = K=2h+1.
// B  4x16 layout: lane -> N, VGPR r holds K = r + 2h.
// C 16x16 layout: VGPR r -> M = r + 8h, lane -> N.
// ---------------------------------------------------------------------------
__global__ __launch_bounds__(256) void gemm0_kernel(const float* __restrict__ X,
                                                    const float* __restrict__ W0,
                                                    const float* __restrict__ b0,
                                                    const u8* __restrict__ adj,
                                                    const int* __restrict__ nl_idx,
                                                    float* __restrict__ stateg) {
    const int wave = blockIdx.x * (blockDim.x >> 5) + (threadIdx.x >> 5);
    const int lane = threadIdx.x & 31;
    const int tm = wave & 3;           // 4 M tiles (64 rows)
    const int tn = wave >> 2;          // 128 N tiles (2048 cols)
    const int m = tm * 16, n0 = tn * 16;
    const int h = lane >> 4;           // lane half (0/1)
    const int r = lane & 15;           // within-half index

    v8f acc = {0.f, 0.f, 0.f, 0.f, 0.f, 0.f, 0.f, 0.f};
    #pragma unroll 4
    for (int k = 0; k < IN_DIM; k += 4) {
        const v2f a = *(const v2f*)(X + (m + r) * IN_DIM + k + 2 * h);
        v2f b;
        b.x = W0[(k + 2 * h)     * NN + n0 + r];
        b.y = W0[(k + 2 * h + 1) * NN + n0 + r];
        acc = __builtin_amdgcn_wmma_f32_16x16x4_f32(false, a, false, b,
                                                    (short)0, acc, false, false);
    }

    const int nl0 = nl_idx[0];               // scalar broadcast
    const int col = n0 + r;
    const float bias = b0[col];
    const float msk = adj[col] ? 1.f : 0.f;  // adj row 0
    #pragma unroll
    for (int j = 0; j < 8; ++j) {
        const int row = m + j + 8 * h;
        stateg[row * NN + col] = msk * apply_nl(nl0, acc[j] + bias);
    }
}

// ---------------------------------------------------------------------------
// Async copy of one 8 KB weight row (2048 f32) into LDS via the CDNA5
// Tensor/async path: GLOBAL_LOAD_ASYNC_TO_LDS_B128, tracked by ASYNCcnt.
// One wave issues 16 instructions (32 lanes x 16 B each = 512 B per instr).
// ---------------------------------------------------------------------------
__device__ __forceinline__ void issue_async_row(const float* __restrict__ gsrc,
                                                unsigned lds_dst, int lane) {
    const char* g = (const char*)gsrc + lane * 16;
    unsigned     l = lds_dst + lane * 16;
    #pragma unroll
    for (int i = 0; i < 16; ++i) {
        asm volatile("global_load_async_to_lds_b128 %0, %1, off"
                     :: "v"(l), "v"(g) : "memory");
        g += 512;
        l += 512;
    }
}

// ---------------------------------------------------------------------------
// Kernel 4: sequential scan over nodes 1..N-2.
// Batch rows are independent -> 16 workgroups x 4 batch rows, one wave32 per
// row, state slice resident in LDS (32 KB). Weight rows are staged ONCE per
// workgroup into a 4-deep LDS ring by wave 0 using async global->LDS copies
// (3-row lookahead hides L2 latency; s_wait_asynccnt throttles precisely).
// ---------------------------------------------------------------------------
__global__ __launch_bounds__(128) void scan_kernel(float* __restrict__ stateg,
                                                   const float* __restrict__ Wt,
                                                   const float* __restrict__ b_mid,
                                                   const int* __restrict__ nl_idx) {
    __shared__ __align__(16) float st[4][NN];   // 32 KB state slice
    __shared__ __align__(16) float wb[4][NN];   // 32 KB weight-row ring
    const int w    = threadIdx.x >> 5;          // wave id == local batch row
    const int lane = threadIdx.x & 31;
    const int b0   = blockIdx.x * 4;

    // init: load state slice, zero the ring (defined contents for the compiler)
    for (int i = threadIdx.x; i < 4 * NN; i += 128) {
        st[i >> 11][i & (NN - 1)] = stateg[(b0 + (i >> 11)) * NN + (i & (NN - 1))];
        ((float*)wb)[i] = 0.f;
    }
    __syncthreads();

    const unsigned lds0 = (unsigned)(size_t)&wb[0][0];  // wave-relative LDS offset

    // prologue: stage rows 1..3 into ring slots 1..3; ensure row 1 landed
    if (w == 0) {
        issue_async_row(Wt + 1 * NN, lds0 + 1 * (NN * 4), lane);
        issue_async_row(Wt + 2 * NN, lds0 + 2 * (NN * 4), lane);
        issue_async_row(Wt + 3 * NN, lds0 + 3 * (NN * 4), lane);
        asm volatile("s_wait_asynccnt 0x20" ::: "memory");  // rows 2,3 may fly
    }
    __syncthreads();

    float* row = st[w];
    for (int n = 1; n <= NN - 2; ++n) {
        // stage row n+3 while computing row n
        if (w == 0 && n + 3 <= NN - 2)
            issue_async_row(Wt + (n + 3) * NN, lds0 + ((n + 3) & 3) * (NN * 4), lane);

        const float* wr = &wb[n & 3][0];
        float acc = 0.f;
        #pragma unroll
        for (int j = 0; j < 16; ++j) {
            const int k = j * 128 + lane * 4;
            const float4 wv = *(const float4*)(wr + k);
            const float4 sv = *(const float4*)(row + k);
            acc = fmaf(wv.x, sv.x, acc);
            acc = fmaf(wv.y, sv.y, acc);
            acc = fmaf(wv.z, sv.z, acc);
            acc = fmaf(wv.w, sv.w, acc);
        }
        #pragma unroll
        for (int off = 16; off > 0; off >>= 1)
            acc += __shfl_xor(acc, off, 32);     // wave32 butterfly
        if (lane == 0)
            row[n] = apply_nl(nl_idx[n], acc + b_mid[n]);

        // guarantee row n+1 is resident before the next step (async loads
        // complete in order; allow rows n+2/n+3 to remain in flight)
        if (w == 0) {
            if      (n + 3 <= NN - 2) asm volatile("s_wait_asynccnt 0x20" ::: "memory");
            else if (n + 2 <= NN - 2) asm volatile("s_wait_asynccnt 0x10" ::: "memory");
            else if (n + 1 <= NN - 2) asm volatile("s_wait_asynccnt 0x0"  ::: "memory");
        }
        __syncthreads();
    }

    for (int i = threadIdx.x; i < 4 * NN; i += 128)
        stateg[(b0 + (i >> 11)) * NN + (i & (NN - 1))] = st[i >> 11][i & (NN - 1)];
}

// ---------------------------------------------------------------------------
// Kernel 5: out = nl_{N-1}(state @ (W_out * mcol[:,None]) + b_out)
// Same f32 WMMA tiling, M=64, N=512, K=2048.
// ---------------------------------------------------------------------------
__global__ __launch_bounds__(256) void gemmout_kernel(const float* __restrict__ stateg,
                                                      const float* __restrict__ W_out,
                                                      const float* __restrict__ b_out,
                                                      const float* __restrict__ mcol,
                                                      const int* __restrict__ nl_idx,
                                                      float* __restrict__ outg) {
    const int wave = blockIdx.x * (blockDim.x >> 5) + (threadIdx.x >> 5);
    const int lane = threadIdx.x & 31;
    const int tm = wave & 3;           // 4 M tiles
    const int tn = wave >> 2;          // 32 N tiles (512 cols)
    const int m = tm * 16, n0 = tn * 16;
    const int h = lane >> 4;
    const int r = lane & 15;

    v8f acc = {0.f, 0.f, 0.f, 0.f, 0.f, 0.f, 0.f, 0.f};
    #pragma unroll 4
    for (int k = 0; k < NN; k += 4) {
        const v2f a = *(const v2f*)(stateg + (m + r) * NN + k + 2 * h);
        const int kb = k + 2 * h;
        v2f b;
        b.x = W_out[kb * OUT_DIM + n0 + r]       * mcol[kb];
        b.y = W_out[(kb + 1) * OUT_DIM + n0 + r] * mcol[kb + 1];
        acc = __builtin_amdgcn_wmma_f32_16x16x4_f32(false, a, false, b,
                                                    (short)0, acc, false, false);
    }

    const int nlN = nl_idx[NN - 1];
    const int col = n0 + r;
    const float bias = b_out[col];
    #pragma unroll
    for (int j = 0; j < 8; ++j) {
        const int row = m + j + 8 * h;
        outg[row * OUT_DIM + col] = apply_nl(nlN, acc[j] + bias);
    }
}

// ---------------------------------------------------------------------------
// Host-side launcher. Inputs (setup_inputs order):
//   0:X f32[64,1024]  1:adj bool[2048,2048] (1 byte/elem)  2:nl_idx i32[2048]
//   3:W0 f32[1024,2048] 4:b0 f32[2048] 5:W_mid f32[2048,2048] 6:b_mid f32[2048]
//   7:W_out f32[2048,512] 8:b_out f32[512]
// d_out = [ out f32[64,512] | state f32[64,2048] ]   (163840 floats)
// d_ws  needs 2048*2048*4 + 2048*4 = 16 MB + 8 KB for Wt and mcol.
// ---------------------------------------------------------------------------
extern "C" void kernel_launch(void* const* d_in, const int* in_sizes, int n_in,
                              void* d_out, int out_size, void* d_ws, size_t ws_size,
                              hipStream_t stream) {
    const float* X      = (const float*)d_in[0];
    const u8*    adj    = (const u8*)   d_in[1];
    const int*   nl_idx = (const int*)  d_in[2];
    const float* W0     = (const float*)d_in[3];
    const float* b0     = (const float*)d_in[4];
    const float* W_mid  = (const float*)d_in[5];
    const float* b_mid  = (const float*)d_in[6];
    const float* W_out  = (const float*)d_in[7];
    const float* b_out  = (const float*)d_in[8];

    float* out    = (float*)d_out;
    float* stateg = out + B_SZ * OUT_DIM;         // state region of d_out
    float* Wt     = (float*)d_ws;                 // 2048*2048 f32
    float* mcol   = Wt + NN * NN;                 // 2048 f32

    // 1) masked transpose of W_mid (one-time; leaves Wt hot in L2)
    tmask_kernel<<<dim3(NN / 32, NN / 32), dim3(32, 8), 0, stream>>>(W_mid, adj, Wt);
    // 2) output-GEMM column mask
    maskcol_kernel<<<NN / 256, 256, 0, stream>>>(adj, mcol);
    // 3) init state = mask * nl0(X@W0 + b0): 4x128 tiles, 8 waves/block
    gemm0_kernel<<<64, 256, 0, stream>>>(X, W0, b0, adj, nl_idx, stateg);
    // 4) sequential node scan, batch-parallel across 16 WGPs, async-staged weights
    scan_kernel<<<B_SZ / 4, 128, 0, stream>>>(stateg, Wt, b_mid, nl_idx);
    // 5) out = nlN(state @ masked W_out + b_out): 4x32 tiles, 8 waves/block
    gemmout_kernel<<<16, 256, 0, stream>>>(stateg, W_out, b_out, mcol, nl_idx, out);
}